// UniformNeighborSampler_85512798863388
// MI455X (gfx1250) — compile-verified
//
#include <hip/hip_runtime.h>
#include <hip/hip_bf16.h>

// ---------------------------------------------------------------------------
// UniformNeighborSampler for MI455X (gfx1250, wave32).
// out[b, s] = adj[g_id, ids[b], sel[s]]  with sel = jax threefry perm(128)[:32]
//
// Kernel 1 (perm_kernel): reproduce JAX's partitionable Threefry-2x32
//   permutation of 128 and write sel[] (num_samples ints) into d_ws.
// Kernel 2 (gather_kernel): one wave32 per batch row. 32 lanes issue one
//   GLOBAL_LOAD_ASYNC_TO_LDS_B128 each -> the whole 512B adjacency row is
//   DMA'd memory->LDS with no VGPR round trip (ASYNCcnt-tracked, CDNA5-only
//   path). After s_wait_asynccnt 0, each lane gathers its selected column
//   from LDS (ds_load_b32) and does a coalesced b32 store. Pure bandwidth:
//   ~32 MB traffic -> ~1.4 us at 23.3 TB/s; 51 MB graph slice is L2-resident.
// ---------------------------------------------------------------------------

#define DEG 128
#define N_GRAPHS 4
#define WAVES_PER_BLOCK 8

__device__ __forceinline__ void threefry2x32(unsigned k0, unsigned k1,
                                             unsigned x0, unsigned x1,
                                             unsigned& o0, unsigned& o1) {
  const unsigned ks2 = k0 ^ k1 ^ 0x1BD11BDAu;
  unsigned v0 = x0 + k0;
  unsigned v1 = x1 + k1;
#define TF_ROT(v, d) (((v) << (d)) | ((v) >> (32 - (d))))
#define TF_R4(a, b, c, d)                                   \
  v0 += v1; v1 = TF_ROT(v1, a); v1 ^= v0;                   \
  v0 += v1; v1 = TF_ROT(v1, b); v1 ^= v0;                   \
  v0 += v1; v1 = TF_ROT(v1, c); v1 ^= v0;                   \
  v0 += v1; v1 = TF_ROT(v1, d); v1 ^= v0;
  TF_R4(13, 15, 26, 6);  v0 += k1;  v1 += ks2 + 1u;
  TF_R4(17, 29, 16, 24); v0 += ks2; v1 += k0  + 2u;
  TF_R4(13, 15, 26, 6);  v0 += k0;  v1 += k1  + 3u;
  TF_R4(17, 29, 16, 24); v0 += k1;  v1 += ks2 + 4u;
  TF_R4(13, 15, 26, 6);  v0 += ks2; v1 += k0  + 5u;
#undef TF_R4
#undef TF_ROT
  o0 = v0;
  o1 = v1;
}

// One block of 128 threads: compute sort keys, stable rank-sort, emit sel[].
__global__ void UniformNeighborSampler_perm_kernel(int* __restrict__ sel,
                                                   int ns) {
  __shared__ unsigned keys[DEG];
  const int t = threadIdx.x;  // 0..127

  // subkey = split(key(42))[1]  (partitionable: counter pair (0, 1))
  unsigned s0, s1;
  threefry2x32(0u, 42u, 0u, 1u, s0, s1);

  // sort_keys[t] = xor of threefry2x32(subkey, (0, t)) outputs
  unsigned o0, o1;
  threefry2x32(s0, s1, 0u, (unsigned)t, o0, o1);
  keys[t] = o0 ^ o1;
  __syncthreads();

  // Stable rank: position of element t in ascending sort of keys.
  const unsigned k = keys[t];
  int rank = 0;
#pragma unroll 8
  for (int j = 0; j < DEG; ++j) {
    const unsigned kj = keys[j];
    rank += (int)((kj < k) | ((kj == k) & (j < t)));
  }
  if (rank < ns) sel[rank] = t;  // perm[rank] = t ; keep first ns
}

// One wave32 per batch row.
__global__ __launch_bounds__(WAVES_PER_BLOCK * 32) void
UniformNeighborSampler_gather_kernel(const int* __restrict__ adj_ins,
                                     const int* __restrict__ adj_outs,
                                     const int* __restrict__ g_id_p,
                                     const int* __restrict__ ins_p,
                                     const int* __restrict__ ids,
                                     const int* __restrict__ sel,
                                     int* __restrict__ out,
                                     int batch,
                                     long long graph_stride,
                                     int ns) {
  __shared__ __align__(16) int rows[WAVES_PER_BLOCK][DEG];

  const int lane = threadIdx.x & 31;
  const int wave = threadIdx.x >> 5;
  const int b = blockIdx.x * WAVES_PER_BLOCK + wave;

  // Uniform scalars (device memory; select table on-device for capture safety)
  const int g = *g_id_p;
  const int insv = *ins_p;
  const int* __restrict__ adj =
      (insv ? adj_ins : adj_outs) + (long long)g * graph_stride;

  if (b < batch) {
    const int row = ids[b];
    // Whole 512B row DMA'd straight into LDS: lane l moves bytes 16l..16l+15.
    // GV mode (SADDR=off): 64-bit global address in a VGPR pair, LDS byte
    // offset (AS(3) offset = low 32 bits of the generic pointer) in VDST.
    const unsigned long long gaddr =
        (unsigned long long)(uintptr_t)((const char*)(adj +
                                                      (long long)row * DEG) +
                                        lane * 16);
    const unsigned lds_addr =
        (unsigned)(uintptr_t)&rows[wave][0] + (unsigned)(lane * 16);
    asm volatile("global_load_async_to_lds_b128 %0, %1, off"
                 :
                 : "v"(lds_addr), "v"(gaddr)
                 : "memory");
  }

  // Drain the async memory->LDS transfers before the cross-lane LDS gather.
  __builtin_amdgcn_wave_barrier();
  asm volatile("s_wait_asynccnt 0x0" ::: "memory");

  if (b < batch) {
    const long long obase = (long long)b * ns;
    for (int s = lane; s < ns; s += 32) {
      const int c = sel[s];              // 32 consecutive ints, L2/L0 resident
      out[obase + s] = rows[wave][c];    // ds_load_b32 + coalesced b32 store
    }
  }
}

extern "C" void kernel_launch(void* const* d_in, const int* in_sizes, int n_in,
                              void* d_out, int out_size, void* d_ws,
                              size_t ws_size, hipStream_t stream) {
  const int* adj_ins  = (const int*)d_in[0];
  const int* adj_outs = (const int*)d_in[1];
  const int* g_id_p   = (const int*)d_in[2];
  const int* ids      = (const int*)d_in[3];
  // d_in[4] = num_samples (derived below from out_size), d_in[5] = ins flag
  const int* ins_p    = (const int*)d_in[5];

  const int batch = in_sizes[3];
  const int ns = out_size / batch;                       // 32
  const long long graph_stride = (long long)in_sizes[0] / N_GRAPHS;  // nodes*DEG

  int* sel = (int*)d_ws;   // ns ints of scratch
  int* out = (int*)d_out;  // int32 written through int* (bit-identical)

  UniformNeighborSampler_perm_kernel<<<1, DEG, 0, stream>>>(sel, ns);

  const int blocks = (batch + WAVES_PER_BLOCK - 1) / WAVES_PER_BLOCK;  // 6250
  UniformNeighborSampler_gather_kernel<<<blocks, WAVES_PER_BLOCK * 32, 0,
                                         stream>>>(
      adj_ins, adj_outs, g_id_p, ins_p, ids, sel, out, batch, graph_stride, ns);
}